// InstanceColouringSBP_9234179687398
// MI455X (gfx1250) — compile-verified
//
#include <hip/hip_runtime.h>
#include <math.h>

typedef __attribute__((ext_vector_type(2))) float v2f;
typedef __attribute__((ext_vector_type(8))) float v8f;

#define KSTEPS   8
#define BATCH    8
#define IMG      256
#define HW       (IMG * IMG)          // 65536
#define FEAT     64
#define CDIM     8
#define NB1      64                   // argmax stage-1 blocks per batch image

// ---------------------------------------------------------------------------
// Phase 1: colour head  colour[b,c,p] = sum_f feat[b,f,p]*W[c,f] + bias[c]
// WMMA f32 16x16x4: A = W tile [16ch x 4feat], B = features [4feat x 16pix],
// C/D = [16ch x 16pix] f32 accumulator. 16 K-steps cover FEAT=64.
// Each wave produces 4 tiles of 16 pixels = 64 pixels.
// Grid is sized exactly: no divergence before WMMA (EXEC must be all-ones).
// ---------------------------------------------------------------------------
__global__ __launch_bounds__(256) void colour_wmma_kernel(
    const float* __restrict__ feat, const float* __restrict__ Wm,
    const float* __restrict__ bias, float* __restrict__ colour)
{
    const int lane  = threadIdx.x & 31;
    const int wave  = blockIdx.x * (blockDim.x >> 5) + (threadIdx.x >> 5);
    const long pix0 = (long)wave * 64;            // global pixel over B*HW
    const int  b    = (int)(pix0 / HW);
    const int  pbase = (int)(pix0 - (long)b * HW);

    // A-matrix (16x4 f32) layout: lanes 0-15 hold K={0,1} in VGPR{0,1},
    // lanes 16-31 hold K={2,3}.  M = lane&15 = channel (ch>=8 padded w/ 0).
    const int c     = lane & 15;
    const int khalf = (lane >> 4) << 1;           // 0 or 2

    v2f a[16];
#pragma unroll
    for (int s = 0; s < 16; ++s) {
        const int k0 = s * 4 + khalf;
        a[s].x = (c < CDIM) ? Wm[c * FEAT + k0]     : 0.0f;
        a[s].y = (c < CDIM) ? Wm[c * FEAT + k0 + 1] : 0.0f;
    }

    const float* fb = feat + (long)b * FEAT * HW;
    const int n = lane & 15;                      // pixel within tile

#pragma unroll
    for (int t = 0; t < 4; ++t) {
        const int p = pbase + t * 16 + n;
        v8f acc = {};
#pragma unroll
        for (int s = 0; s < 16; ++s) {
            const int k0 = s * 4 + khalf;
            v2f bv;
            bv.x = fb[(long)k0       * HW + p];
            bv.y = fb[(long)(k0 + 1) * HW + p];
            acc = __builtin_amdgcn_wmma_f32_16x16x4_f32(
                false, a[s], false, bv, (short)0, acc, false, false);
        }
        // D layout: VGPR r, lanes 0-15 -> channel r (lanes 16-31 -> ch r+8,
        // which is padding).  Contiguous 16-float store per channel row.
        if (lane < 16) {
            float* cb = colour + (long)b * CDIM * HW;
            const int pp = pbase + t * 16 + lane;
#pragma unroll
            for (int r = 0; r < CDIM; ++r)
                cb[(long)r * HW + pp] = acc[r] + bias[r];
        }
    }
}

// ---------------------------------------------------------------------------
// log_s_k[0] = 0
// ---------------------------------------------------------------------------
__global__ __launch_bounds__(256) void init_zero_kernel(float* __restrict__ dst)
{
    const long i = (long)blockIdx.x * blockDim.x + threadIdx.x;
    dst[i] = 0.0f;
}

// ---------------------------------------------------------------------------
// Scan step, stage 1: per-batch partial argmax of rand * exp(log_s).
// Tie-break: lowest index (matches jnp.argmax first-occurrence).
// ---------------------------------------------------------------------------
__global__ __launch_bounds__(256) void argmax_part_kernel(
    const float* __restrict__ rand_pixel, const float* __restrict__ log_s,
    float* __restrict__ pvals, int* __restrict__ pidx)
{
    const int b     = blockIdx.x / NB1;
    const int chunk = blockIdx.x % NB1;
    const int base  = chunk * (HW / NB1);
    const float* rp = rand_pixel + (long)b * HW;
    const float* ls = log_s      + (long)b * HW;

    float best = -1.0f;
    int   bidx = 0x7fffffff;
    for (int i = threadIdx.x; i < HW / NB1; i += 256) {
        const int p = base + i;
        const float v = rp[p] * expf(ls[p]);
        if (v > best || (v == best && p < bidx)) { best = v; bidx = p; }
    }

    __shared__ float sv[256];
    __shared__ int   si[256];
    sv[threadIdx.x] = best; si[threadIdx.x] = bidx;
    __syncthreads();
    for (int off = 128; off > 0; off >>= 1) {
        if ((int)threadIdx.x < off) {
            const float ov = sv[threadIdx.x + off];
            const int   oi = si[threadIdx.x + off];
            if (ov > sv[threadIdx.x] ||
                (ov == sv[threadIdx.x] && oi < si[threadIdx.x])) {
                sv[threadIdx.x] = ov; si[threadIdx.x] = oi;
            }
        }
        __syncthreads();
    }
    if (threadIdx.x == 0) { pvals[blockIdx.x] = sv[0]; pidx[blockIdx.x] = si[0]; }
}

// ---------------------------------------------------------------------------
// Stage 2: reduce NB1 partials per batch -> idx; gather seed colour.
// ---------------------------------------------------------------------------
__global__ __launch_bounds__(NB1) void argmax_final_kernel(
    const float* __restrict__ pvals, const int* __restrict__ pidx,
    const float* __restrict__ colour, float* __restrict__ seed_ws,
    float* __restrict__ seeds_out)
{
    const int b = blockIdx.x;
    const int t = threadIdx.x;
    __shared__ float sv[NB1];
    __shared__ int   si[NB1];
    sv[t] = pvals[b * NB1 + t]; si[t] = pidx[b * NB1 + t];
    __syncthreads();
    for (int off = NB1 / 2; off > 0; off >>= 1) {
        if (t < off) {
            const float ov = sv[t + off];
            const int   oi = si[t + off];
            if (ov > sv[t] || (ov == sv[t] && oi < si[t])) {
                sv[t] = ov; si[t] = oi;
            }
        }
        __syncthreads();
    }
    if (t < CDIM) {
        const int idx = si[0];
        const float s = colour[(long)b * CDIM * HW + (long)t * HW + idx];
        seed_ws[b * CDIM + t]   = s;   // for the map kernel
        seeds_out[b * CDIM + t] = s;   // output seeds[k,b,c]
    }
}

// ---------------------------------------------------------------------------
// Scan step map: per pixel
//   dist  = sum_c (colour - seed)^2
//   alpha = clamp(exp(-dist/sigma), 0.01, 0.99)
//   log_m_k[k] = log_s + log(alpha)
//   log_s_k[k+1] = log_s + log(1-alpha)   (and log_m_k[K] on last step)
// ---------------------------------------------------------------------------
__global__ __launch_bounds__(256) void scan_map_kernel(
    const float* __restrict__ colour, const float* __restrict__ seed_ws,
    const float* __restrict__ log_sigma_ptr,
    const float* __restrict__ log_s_in,
    float* __restrict__ log_s_out,
    float* __restrict__ log_m_out,
    float* __restrict__ log_m_last,
    int is_last)
{
    const long i = (long)blockIdx.x * blockDim.x + threadIdx.x;  // over B*HW
    const int  b = (int)(i / HW);
    const int  p = (int)(i - (long)b * HW);
    const float sigma = expf(log_sigma_ptr[0]);

    const float* cb = colour + (long)b * CDIM * HW + p;
    float dist = 0.0f;
#pragma unroll
    for (int cc = 0; cc < CDIM; ++cc) {
        const float d = cb[(long)cc * HW] - seed_ws[b * CDIM + cc];
        dist += d * d;
    }
    float alpha = expf(-dist / sigma);
    alpha = fminf(fmaxf(alpha, 0.01f), 0.99f);

    const float ls  = log_s_in[i];
    log_m_out[i]    = ls + logf(alpha);
    const float nls = ls + logf(1.0f - alpha);
    log_s_out[i]    = nls;
    if (is_last) log_m_last[i] = nls;
}

// ---------------------------------------------------------------------------
extern "C" void kernel_launch(void* const* d_in, const int* in_sizes, int n_in,
                              void* d_out, int out_size, void* d_ws, size_t ws_size,
                              hipStream_t stream)
{
    const float* feat       = (const float*)d_in[0];  // [B,F,H,W]
    const float* Wm         = (const float*)d_in[1];  // [C,F]
    const float* bias       = (const float*)d_in[2];  // [C]
    const float* log_sigma  = (const float*)d_in[3];  // scalar
    const float* rand_pixel = (const float*)d_in[4];  // [B,1,H,W]
    // d_in[5] = steps_to_run (8; compile-time — output geometry depends on it)

    float* out = (float*)d_out;
    const long slice = (long)BATCH * HW;                  // one [B,1,H,W] slab
    float* log_m  = out;                                  // [K+1,B,1,H,W]
    float* log_s  = out + (long)(KSTEPS + 1) * slice;     // [K+1,B,1,H,W]
    float* colour = out + 2L * (KSTEPS + 1) * slice;      // [B,C,H,W]
    float* seeds  = colour + (long)BATCH * CDIM * HW;     // [K,B,C]

    // workspace: argmax partials + per-step seeds
    float* pvals   = (float*)d_ws;                        // B*NB1
    int*   pidx    = (int*)(pvals + BATCH * NB1);         // B*NB1
    float* seed_ws = (float*)(pidx + BATCH * NB1);        // B*CDIM

    // Phase 1: WMMA colour head.  8192 waves * 64 pixels == B*HW exactly.
    colour_wmma_kernel<<<1024, 256, 0, stream>>>(feat, Wm, bias, colour);

    // log_s_k[0] = 0
    init_zero_kernel<<<(int)(slice / 256), 256, 0, stream>>>(log_s);

    // Phase 2: sequential scan, 8 steps.
    for (int k = 0; k < KSTEPS; ++k) {
        argmax_part_kernel<<<BATCH * NB1, 256, 0, stream>>>(
            rand_pixel, log_s + (long)k * slice, pvals, pidx);
        argmax_final_kernel<<<BATCH, NB1, 0, stream>>>(
            pvals, pidx, colour, seed_ws, seeds + (long)k * BATCH * CDIM);
        scan_map_kernel<<<(int)(slice / 256), 256, 0, stream>>>(
            colour, seed_ws, log_sigma,
            log_s + (long)k * slice,
            log_s + (long)(k + 1) * slice,
            log_m + (long)k * slice,
            log_m + (long)KSTEPS * slice,
            (k == KSTEPS - 1) ? 1 : 0);
    }
}